// ProdEinsumTC_17600775979374
// MI455X (gfx1250) — compile-verified
//
#include <hip/hip_runtime.h>
#include <stdint.h>

// ---------------------------------------------------------------------------
// out[a,bc,d,g] = (1/3) * sum_k T1[a,bc,d,k] * T2[bc,k,g], k=(e,f) in 0..8
//   a=4, bc=786432, d,g in 0..2.  ~540 MB moved, ~0.5 GFLOP -> HBM-bound
//   (floor ~23us at 23.3 TB/s).  f32 kept end-to-end.
// Strategy:
//   * WG of 256 threads stages 40 problems of T2 + 4x40 of T1 into LDS with
//     gfx1250 async global->LDS b128 copies (ASYNCcnt), branch-free.
//   * Each wave packs 5 problems block-diagonally in a 16x16 tile and runs
//     three chained V_WMMA_F32_16X16X4_F32 (K=9 zero-padded to 12).
//     Pad row/col 15 carries garbage that only lands in C row/col 15 (never
//     read); only K>=9 needs explicit zeroing (one cndmask).
//   * B (T2) fragments built once per tile, reused across the 4 paths.
//   * C spilled lane-contiguously (2x ds_store_b128); 45 outputs per wave
//     written back coalesced with 32-bit offset addressing.
// ---------------------------------------------------------------------------

#define PATH      4
#define BC        786432            // 131072 * 6 fused (batch, channel)
#define TPP       27                // floats per problem in T1/T2 (3*3*3)
#define OPP       9                 // outputs per problem (3*3)
#define NORM      (1.0f / 3.0f)

#define PER_WAVE  5                 // problems packed into one 16x16 tile
#define WAVES     8
#define PER_WG    (PER_WAVE * WAVES)                 // 40
#define NTILES    ((BC + PER_WG - 1) / PER_WG)       // 19661 (last: 32 valid)

typedef __attribute__((ext_vector_type(2))) float v2f;
typedef __attribute__((ext_vector_type(8))) float v8f;
typedef int v4i_g __attribute__((vector_size(16)));  // GCC vector: builtin proto

#if defined(__HIP_DEVICE_COMPILE__) && __has_builtin(__builtin_amdgcn_global_load_async_to_lds_b128)
#define USE_ASYNC 1
#else
#define USE_ASYNC 0
#endif

#if defined(__HIP_DEVICE_COMPILE__) && __has_builtin(__builtin_amdgcn_wmma_f32_16x16x4_f32)
#define USE_WMMA 1
#else
#define USE_WMMA 0
#endif

__device__ __forceinline__ void stage16(const float* g, float* l) {
#if USE_ASYNC
  __builtin_amdgcn_global_load_async_to_lds_b128(
      (__attribute__((address_space(1))) v4i_g*)(uintptr_t)g,
      (__attribute__((address_space(3))) v4i_g*)(uint32_t)(uintptr_t)l,
      /*offset=*/0, /*cpol=*/0);
#else
  *(float4*)l = *(const float4*)g;
#endif
}

__device__ __forceinline__ void stage_wait() {
#if USE_ASYNC
#if __has_builtin(__builtin_amdgcn_s_wait_asynccnt)
  __builtin_amdgcn_s_wait_asynccnt(0);
#else
  asm volatile("s_wait_asynccnt 0" ::: "memory");
#endif
#endif
}

__device__ __forceinline__ void wave_fence() {
#if defined(__HIP_DEVICE_COMPILE__)
  __builtin_amdgcn_fence(__ATOMIC_ACQ_REL, "wavefront");
#endif
}

__global__ __launch_bounds__(256)
void einsum_tc_kernel(const float* __restrict__ T1,
                      const float* __restrict__ T2,
                      float* __restrict__ out) {
  __shared__ float sT2[PER_WG * TPP];          // 4.32 KB
  __shared__ float sT1[PATH][PER_WG * TPP];    // 17.28 KB
  __shared__ float sC[WAVES][16 * 16];         // 8 KB (lane-contiguous spill)

  const int tid  = threadIdx.x;
  const int wave = tid >> 5;
  const int lane = tid & 31;
  const int half = lane >> 4;                  // 0: lanes 0-15, 1: lanes 16-31
  const int l    = lane & 15;

  const int pbase = (int)blockIdx.x * PER_WG;  // all offsets fit in 32 bits
  const int remp  = BC - pbase;
  const int valid = remp < PER_WG ? remp : PER_WG;
  const int vfl   = valid * TPP;               // 1080 or 864: multiple of 4

  // ---- stage T2 + 4x T1 into LDS: branch-free, 2 predicated b128 steps ----
  {
    const int i0 = tid * 4, i1 = i0 + 1024;    // vfl <= 1080 < 2048
    const float* g2 = T2 + pbase * TPP;
    if (i0 < vfl) stage16(g2 + i0, &sT2[i0]);
    if (i1 < vfl) stage16(g2 + i1, &sT2[i1]);
#pragma unroll
    for (int a = 0; a < PATH; ++a) {
      const float* g1 = T1 + (a * BC + pbase) * TPP;
      if (i0 < vfl) stage16(g1 + i0, &sT1[a][i0]);
      if (i1 < vfl) stage16(g1 + i1, &sT1[a][i1]);
    }
    stage_wait();
  }
  __syncthreads();

  // Tile mapping: M = 3p+d (rows, A/C), N = 3p+g (cols, B/C), p = 0..4.
  const int w5  = wave * PER_WAVE;
  const int pl  = l / 3;
  const int el  = l - pl * 3;                  // d (rows) / g (cols)
  const int pls = (pl < PER_WAVE) ? pl : 0;    // in-bounds index for pad lane
  const int pbP = (w5 + pls) * TPP;
  const int h2  = half * 2;

#if USE_WMMA
  // ---- B fragments (shared across the 4 paths). B[k][n] = T2[p_n][k][g_n].
  // K striping: VGPR0 holds k = kc*4 + 2*half, VGPR1 holds k+1.
  const float* sB = &sT2[pbP + el];
  v2f bf0, bf1, bf2;
  bf0[0] = sB[(h2 + 0) * 3];
  bf0[1] = sB[(h2 + 1) * 3];
  bf1[0] = sB[(h2 + 4) * 3];
  bf1[1] = sB[(h2 + 5) * 3];
  {
    const float t = sB[24];                    // k = 8
    bf2[0] = half ? 0.0f : t;                  // k=10 is pad -> 0
    bf2[1] = 0.0f;                             // k=9 / k=11 are pad -> 0
  }
#endif

  // ---- hoisted output decode (independent of a) ---------------------------
  // element j of this wave's 45-float segment: p=j/9, d=(j%9)/3, g=j%3
  // C[m][n] lives at sC[wave][(n + 16*(m>>3))*8 + (m&7)]
  const int j0 = lane;                         // 0..31 (< 45 always)
  const int p0 = j0 / 9, r90 = j0 - p0 * 9;
  const int d0 = r90 / 3, g0 = r90 - d0 * 3;
  const int m0 = 3 * p0 + d0, n0 = 3 * p0 + g0;
  const int ci0 = (n0 + ((m0 >> 3) << 4)) * 8 + (m0 & 7);
  const bool ok0 = (pbase + w5 + p0) < BC;

  const int j1 = (lane < 13) ? lane + 32 : 0;  // clamped: decode stays in-bounds
  const int p1 = j1 / 9, r91 = j1 - p1 * 9;
  const int d1 = r91 / 3, g1 = r91 - d1 * 3;
  const int m1 = 3 * p1 + d1, n1 = 3 * p1 + g1;
  const int ci1 = (n1 + ((m1 >> 3) << 4)) * 8 + (m1 & 7);
  const bool ok1 = (lane < 13) && ((pbase + w5 + p1) < BC);

  const int ob = (pbase + w5) * OPP;           // per-path segment base (32-bit)
  float* cw = &sC[wave][lane * 8];
  const float* cr = &sC[wave][0];

#pragma unroll
  for (int a = 0; a < PATH; ++a) {
#if USE_WMMA
    const float* sAr = &sT1[a][pbP + el * 9];  // A row: T1[a][p_m][d_m][k]
    v2f af0, af1, af2;
    af0[0] = sAr[h2 + 0];
    af0[1] = sAr[h2 + 1];
    af1[0] = sAr[h2 + 4];
    af1[1] = sAr[h2 + 5];
    {
      const float t = sAr[8];
      af2[0] = half ? 0.0f : t;
      af2[1] = 0.0f;
    }
    v8f c = {0.0f, 0.0f, 0.0f, 0.0f, 0.0f, 0.0f, 0.0f, 0.0f};
    c = __builtin_amdgcn_wmma_f32_16x16x4_f32(false, af0, false, bf0, (short)0, c, false, false);
    c = __builtin_amdgcn_wmma_f32_16x16x4_f32(false, af1, false, bf1, (short)0, c, false, false);
    c = __builtin_amdgcn_wmma_f32_16x16x4_f32(false, af2, false, bf2, (short)0, c, false, false);
    // lane-contiguous spill: C[r + 8*half][l] -> sC[wave][lane*8 + r]
#pragma unroll
    for (int r = 0; r < 8; ++r) cw[r] = c[r];
#else
    // scalar fallback producing the identical spill layout
    {
      float bcol[9];
#pragma unroll
      for (int k = 0; k < 9; ++k) bcol[k] = sT2[pbP + k * 3 + el];
#pragma unroll
      for (int r = 0; r < 8; ++r) {
        const int m = r + 8 * half;
        float acc = 0.0f;
        if (m < 15 && l < 15) {
          const int pm = m / 3, dm = m - (m / 3) * 3;
          const float* arow = &sT1[a][(w5 + pm) * TPP + dm * 9];
#pragma unroll
          for (int k = 0; k < 9; ++k) acc += arow[k] * bcol[k];
        }
        cw[r] = acc;
      }
    }
#endif
    wave_fence();

    // ---- coalesced writeback: 32-bit offsets off the SGPR out base --------
    float* opa = out + (a * (BC * OPP) + ob);
    if (ok0) opa[j0] = cr[ci0] * NORM;
    if (ok1) opa[lane + 32] = cr[ci1] * NORM;

    wave_fence();                              // WAR guard before next spill
  }
}

extern "C" void kernel_launch(void* const* d_in, const int* in_sizes, int n_in,
                              void* d_out, int out_size, void* d_ws, size_t ws_size,
                              hipStream_t stream) {
  (void)in_sizes; (void)n_in; (void)out_size; (void)d_ws; (void)ws_size;
  const float* T1 = (const float*)d_in[0];   // (4, 131072, 6, 3, 3, 3) f32
  const float* T2 = (const float*)d_in[1];   // (131072, 6, 3, 3, 3)    f32
  float* out = (float*)d_out;                // (4, 131072, 6, 3, 3)    f32
  einsum_tc_kernel<<<dim3(NTILES), dim3(256), 0, stream>>>(T1, T2, out);
}